// binary_decoder_73555609911822
// MI455X (gfx1250) — compile-verified
//
#include <hip/hip_runtime.h>
#include <stdint.h>

// Problem constants (from reference): B=8, IN_F=4096, OUT_F=512, NBITS=8
#define B_       8
#define IN_F     4096
#define OUT_F    512
#define NBITS    8
#define ROWSTR   (OUT_F * NBITS)        // 4096 floats per weight row
#define TILE     256                    // rows (i-steps) per LDS tile
#define XS       (TILE + 4)             // padded LDS stride for x rows (bank-conflict free, 16B aligned)
#define NTILES   (IN_F / TILE)          // 16

// ---- CDNA5 async global->LDS copy (ASYNCcnt-tracked), per-lane 16B ----
__device__ __forceinline__ void async_copy_b128(uint32_t lds_byte, const void* gptr) {
  uint64_t ga = (uint64_t)(uintptr_t)gptr;
  asm volatile("global_load_async_to_lds_b128 %0, %1, off"
               :: "v"(lds_byte), "v"(ga)
               : "memory");
}

#define WAIT_ASYNCCNT(n) asm volatile("s_wait_asynccnt " #n ::: "memory")

// shift value up by one lane within each 8-lane group (lane%8==0 gets garbage,
// caller masks it). Single v_mov_b32_dpp8 on gfx1250.
__device__ __forceinline__ float shift_up8(float v) {
#if __has_builtin(__builtin_amdgcn_mov_dpp8)
  int i = __builtin_bit_cast(int, v);
  // sel per lane in group of 8: [0,0,1,2,3,4,5,6]
  i = __builtin_amdgcn_mov_dpp8(
      i, (0) | (0 << 3) | (1 << 6) | (2 << 9) | (3 << 12) | (4 << 15) | (5 << 18) | (6 << 21));
  return __builtin_bit_cast(float, i);
#else
  return __shfl_up(v, 1, 8);
#endif
}

// One block per output feature o. 64 threads = 2 waves.
// lane -> (batch = wave*4 + lane>>3, bit = lane&7); each lane owns one
// (batch, o, bit) CSA chain over all 4096 input rows, state in registers.
__global__ void __launch_bounds__(64)
binary_decoder_csa_kernel(const float* __restrict__ xg,
                          const float* __restrict__ wg,
                          float* __restrict__ out) {
  __shared__ __align__(16) float wbuf[2][TILE * NBITS];  // [k*8 + bit]
  __shared__ __align__(16) float xbuf[2][B_ * XS];       // [batch*XS + k]

  const int tid   = threadIdx.x;
  const int lane  = tid & 31;
  const int batch = (tid >> 5) * 4 + (lane >> 3);
  const int bit   = lane & 7;
  const int o     = blockIdx.x;
  const bool carry_en = (bit != 0);

  const char* wcol  = (const char*)(wg + (size_t)o * NBITS);  // weight[:, o*8 .. o*8+7]
  const char* xbase = (const char*)xg;

  // Issue one tile's worth of async copies: 16 b128 instructions per thread.
  auto fill = [&](int tile_base, int buf) {
    // weight tile: 256 rows x 32B  => 512 x 16B chunks over 64 threads
    const char* wt = wcol + (size_t)tile_base * (ROWSTR * 4);
    uint32_t    wl = (uint32_t)(uintptr_t)&wbuf[buf][0];
#pragma unroll
    for (int it = 0; it < 8; ++it) {
      int chunk = it * 64 + tid;       // 0..511
      int row   = chunk >> 1;
      int half  = (chunk & 1) << 4;    // 0 or 16 bytes
      async_copy_b128(wl + (uint32_t)(row * 32 + half),
                      wt + (size_t)row * (ROWSTR * 4) + half);
    }
    // x tile: 8 rows x 1KB => 512 x 16B chunks over 64 threads
    const char* xt = xbase + (size_t)tile_base * 4;
    uint32_t    xl = (uint32_t)(uintptr_t)&xbuf[buf][0];
#pragma unroll
    for (int it = 0; it < 8; ++it) {
      int chunk = it * 64 + tid;       // 0..511
      int bc    = chunk >> 6;          // batch 0..7
      int j     = (chunk & 63) << 4;   // byte offset in row segment
      async_copy_b128(xl + (uint32_t)(bc * (XS * 4) + j),
                      xt + (size_t)bc * (IN_F * 4) + j);
    }
  };

  float s = 0.0f, c = 0.0f;

  fill(0, 0);

  for (int t = 0; t < NTILES; ++t) {
    const int buf = t & 1;
    if (t + 1 < NTILES) {
      fill((t + 1) * TILE, buf ^ 1);   // prefetch next tile into other buffer
      WAIT_ASYNCCNT(16);               // oldest 16 (= current tile) complete
    } else {
      WAIT_ASYNCCNT(0);
    }
    __syncthreads();                   // cross-wave: all fills of buf visible

    const float* wrow = &wbuf[buf][bit];
    const float* xrow = &xbuf[buf][batch * XS];

    for (int k = 0; k < TILE; k += 4) {
      float4 xq = *(const float4*)(xrow + k);
#pragma unroll
      for (int u = 0; u < 4; ++u) {
        float w   = wrow[(k + u) * NBITS];          // ds_load_b32, conflict-free
        float xv  = ((const float*)&xq)[u];
        float xop = (w >= 0.5f) ? xv : 0.0f;        // x * hard
        // soft-logic CSA step:
        //   tt = xor(s,c) = s + c - 2sc
        //   m  = maj(s,c,xop) = sc + xop*tt
        //   ns = xor(tt,xop) = tt + xop - 2*tt*xop
        float p  = s * c;
        float tt = (s + c) - 2.0f * p;
        float m  = __builtin_fmaf(xop, tt, p);
        float ns = (tt + xop) - 2.0f * (tt * xop);
        float mu = shift_up8(m);                    // carry shifts up one bit
        c = carry_en ? mu : 0.0f;                   // bit 0 carry-in = 0
        s = ns;
      }
    }
    __syncthreads();                   // reads of buf done before refill at t+2
  }

  // s reshaped (B*out, nb) -> (B, nb*out) is flat-identical:
  // index = batch*4096 + o*8 + bit; c follows at offset B*OUT_F*NBITS.
  const int base = batch * (OUT_F * NBITS) + o * NBITS + bit;
  out[base] = s;
  out[B_ * OUT_F * NBITS + base] = c;
}

extern "C" void kernel_launch(void* const* d_in, const int* in_sizes, int n_in,
                              void* d_out, int out_size, void* d_ws, size_t ws_size,
                              hipStream_t stream) {
  (void)in_sizes; (void)n_in; (void)d_ws; (void)ws_size; (void)out_size;
  const float* x      = (const float*)d_in[0];
  const float* weight = (const float*)d_in[1];
  float*       out    = (float*)d_out;
  binary_decoder_csa_kernel<<<dim3(OUT_F), dim3(64), 0, stream>>>(x, weight, out);
}